// TimesNet_41918880809321
// MI455X (gfx1250) — compile-verified
//
#include <hip/hip_runtime.h>
#include <stdint.h>

// Problem geometry from the reference setup_inputs():
//   features/mask: [BN, 64, 2048] fp32, valid_lengths: [BN] int32, target_len = 96
constexpr int T_DIM  = 2048;
constexpr int C_DIM  = 64;
constexpr int TS     = 96;             // target steps
constexpr int CPW    = 8;              // channels per wave
constexpr int GROUPS = C_DIM / CPW;    // 8 channel-groups per batch row
constexpr int ROWS   = 2 * CPW;        // 16 row-tasks per wave (features + mask)

// One wave (32 lanes) per block. Streams 16 rows of 8KB through a
// double-buffered LDS staging area using CDNA5 async global->LDS DMA
// (non-temporal: data is read exactly once), then computes 96
// trailing-window adaptive-average-pool outputs per row.
__global__ __launch_bounds__(32) void pool96_kernel(
    const float* __restrict__ feat,
    const float* __restrict__ mask,
    const int*   __restrict__ lens,
    float*       __restrict__ out,
    int BN)
{
    __shared__ __align__(16) float sbuf[2][T_DIM];   // 16 KB double buffer

    const int lane = threadIdx.x;                 // 0..31 (wave32)
    const int blk  = blockIdx.x;
    const int b    = blk / GROUPS;
    const int g    = blk - b * GROUPS;
    const int c0   = g * CPW;

    // L = clip(len, 1, T); trailing window offset = T - L
    int L = lens[b];
    L = L < 1 ? 1 : (L > T_DIM ? T_DIM : L);
    const int off = T_DIM - L;

    // Per-lane segment boundaries for j = lane, lane+32, lane+64.
    // start = floor(L*j/96) + off ; end = ceil(L*(j+1)/96) + off ; count >= 1.
    int segStart[3];
    int segCnt[3];
#pragma unroll
    for (int k = 0; k < 3; ++k) {
        const int j  = lane + 32 * k;
        const int sr = (L * j) / TS;
        const int er = (L * (j + 1) + (TS - 1)) / TS;
        segStart[k] = sr + off;
        int c = er - sr;
        segCnt[k] = c < 1 ? 1 : c;
    }

    const size_t rowBase = (size_t)b * C_DIM;
    const size_t halfOut = (size_t)BN * C_DIM * TS;  // mask outputs follow feature outputs

    // Issue one 8KB row as 16 async b128 copies (32 lanes x 16B each),
    // non-temporal so the one-shot stream doesn't pollute WGP$/L2.
    auto issue_row = [&](int r, int parity) {
        const int c = c0 + (r & (CPW - 1));
        const float* src = ((r < CPW) ? feat : mask) + (rowBase + c) * (size_t)T_DIM;
        // Low 32 bits of the generic shared-aperture address == LDS group-segment
        // byte offset, which is exactly what the async op's VDST expects.
        const uint32_t lds0 = (uint32_t)(uintptr_t)&sbuf[parity][0];
#pragma unroll
        for (int i = 0; i < T_DIM / 128; ++i) {        // 16 issues, 128 floats each
            const uint32_t boff = (uint32_t)((i * 32 + lane) * 16);
            const uint32_t ldsA = lds0 + boff;
            const void* gA = (const void*)((const char*)src + boff);
            asm volatile("global_load_async_to_lds_b128 %0, %1, off th:TH_LOAD_NT"
                         :: "v"(ldsA), "v"(gA)
                         : "memory");
        }
    };

    issue_row(0, 0);

    for (int r = 0; r < ROWS; ++r) {
        const int par = r & 1;
        if (r + 1 < ROWS) {
            issue_row(r + 1, par ^ 1);
            // 32 async ops in flight; async loads retire in order, so waiting
            // down to 16 outstanding guarantees row r is fully in LDS while
            // row r+1 keeps streaming.
            asm volatile("s_wait_asynccnt 0x10" ::: "memory");
        } else {
            asm volatile("s_wait_asynccnt 0x0" ::: "memory");
        }

        const int c = c0 + (r & (CPW - 1));
        float* op = out + ((r < CPW) ? 0 : halfOut) + (rowBase + c) * (size_t)TS;

#pragma unroll
        for (int k = 0; k < 3; ++k) {
            float sum = 0.0f;
            const int st = segStart[k];
            const int n  = segCnt[k];
            for (int t = 0; t < n; ++t) sum += sbuf[par][st + t];
            // Output is written once and never re-read here -> NT store.
            __builtin_nontemporal_store(sum / (float)segCnt[k], &op[lane + 32 * k]);
        }
    }
}

extern "C" void kernel_launch(void* const* d_in, const int* in_sizes, int n_in,
                              void* d_out, int out_size, void* d_ws, size_t ws_size,
                              hipStream_t stream) {
    const float* feat = (const float*)d_in[0];
    const float* mask = (const float*)d_in[1];
    const int*   lens = (const int*)d_in[2];
    float* out = (float*)d_out;

    const int BN = in_sizes[2];          // one length per batch row

    dim3 grid((unsigned)(BN * GROUPS));  // 2048 * 8 = 16384 single-wave blocks
    dim3 block(32);                      // wave32-native
    pool96_kernel<<<grid, block, 0, stream>>>(feat, mask, lens, out, BN);
}